// TLModel_7791070675017
// MI455X (gfx1250) — compile-verified
//
#include <hip/hip_runtime.h>
#include <hip/hip_bf16.h>

typedef float v2f  __attribute__((ext_vector_type(2)));
typedef float v2fu __attribute__((ext_vector_type(2), aligned(4)));  // 4B-aligned vector load
typedef float v8f  __attribute__((ext_vector_type(8)));

// ---------------------------------------------------------------------------
// Layer-1 fusion with padded layout:
//   Weff[o2, c*12+k] = sum_o1 ws[o2,o1,c]*wt[o1,k]   (k<10; k=10,11 -> 0)
//   rows o2 in [25,32) -> 0   (M padded so WMMA A-loads need no masks)
//   beff[o2] = b_spat[o2] + sum_o1 b_time[o1] * sum_c ws[o2,o1,c]
// ---------------------------------------------------------------------------
__global__ void weff_kernel(const float* __restrict__ wt,   // [25,10]
                            const float* __restrict__ bt,   // [25]
                            const float* __restrict__ ws,   // [25,25,22]
                            const float* __restrict__ bs,   // [25]
                            float* __restrict__ weff,       // [32, 22*12]
                            float* __restrict__ beff)       // [25]
{
    int idx = blockIdx.x * blockDim.x + threadIdx.x;
    if (idx < 32 * 264) {
        int o2  = idx / 264;
        int rem = idx - o2 * 264;
        int c   = rem / 12;
        int k   = rem - c * 12;
        float s = 0.f;
        if (o2 < 25 && k < 10)
            for (int o1 = 0; o1 < 25; ++o1)
                s += ws[o2 * 550 + o1 * 22 + c] * wt[o1 * 10 + k];
        weff[idx] = s;
    }
    if (idx < 25) {
        float s = bs[idx];
        for (int o1 = 0; o1 < 25; ++o1) {
            float cs = 0.f;
            for (int c = 0; c < 22; ++c) cs += ws[idx * 550 + o1 * 22 + c];
            s += cs * bt[o1];
        }
        beff[idx] = s;
    }
}

// ---------------------------------------------------------------------------
// Repack OIHW conv weights [M, Cin, 10] -> padded [Mpad, Cin*12] (zero fill).
// ---------------------------------------------------------------------------
__global__ void repack_kernel(const float* __restrict__ src,
                              float* __restrict__ dst,
                              int M, int Mpad, int Cin)
{
    int idx = blockIdx.x * blockDim.x + threadIdx.x;
    int total = Mpad * Cin * 12;
    if (idx >= total) return;
    int m   = idx / (Cin * 12);
    int rem = idx - m * (Cin * 12);
    int c   = rem / 12;
    int t   = rem - c * 12;
    dst[idx] = (m < M && t < 10) ? src[(m * Cin + c) * 10 + t] : 0.f;
}

// ---------------------------------------------------------------------------
// Implicit-GEMM "valid" 10-tap conv via V_WMMA_F32_16X16X4_F32, fp32 exact.
//   in   : [B, Cin, Tin]
//   wpad : [Mpad, Cin*12]   (taps padded 10->12 with zeros, rows padded)
//   out  : [B, M, N], N = Tin-9
// One wave32 computes a 16x32 tile (two n-tiles, shared A fragment).
// Per channel: 3 A b64 loads, 6 B b64 loads, 6 WMMAs -> 1 VMEM : ~1365 FLOP.
// Edge columns are clamped (dup col N-1) and discarded at store, so the hot
// loop has no per-lane masks and EXEC stays all-ones for every WMMA.
// ---------------------------------------------------------------------------
template <int Cin, int Tin, int M>
__global__ __launch_bounds__(32)
void gemm_conv_wmma(const float* __restrict__ in,
                    const float* __restrict__ wpad,
                    const float* __restrict__ bias,
                    float* __restrict__ out)
{
    constexpr int N  = Tin - 9;
    constexpr int Kp = Cin * 12;

    const int lane = threadIdx.x;            // 0..31 (wave32)
    const int half = lane >> 4;              // 0 | 1
    const int l16  = lane & 15;
    const int koff = half * 2;               // K sub-offset: {0,1} vs {2,3}

    const int n0 = blockIdx.x * 32;
    const int m0 = blockIdx.y * 16;
    const int b  = blockIdx.z;

    const float* __restrict__ inb = in + (size_t)b * Cin * Tin;
    const int arow = m0 + l16;                               // < Mpad (padded rows)
    const int bcol0 = n0 + l16;
    const int bcol1 = n0 + 16 + l16;
    const int bc0 = (bcol0 < N) ? bcol0 : (N - 1);           // clamp, no OOB
    const int bc1 = (bcol1 < N) ? bcol1 : (N - 1);

    const float* __restrict__ wrow = wpad + (size_t)arow * Kp + koff;
    const float* __restrict__ p0 = inb + bc0;
    const float* __restrict__ p1 = inb + bc1;

    const v2f z2 = {0.f, 0.f};
    v8f acc0 = {}, acc1 = {};
#pragma unroll 2
    for (int c = 0; c < Cin; ++c) {
        const float* __restrict__ wc = wrow + c * 12;
        const float* __restrict__ i0 = p0 + c * Tin;
        const float* __restrict__ i1 = p1 + c * Tin;
        __builtin_prefetch(i0 + Tin, 0, 1);                  // next channel (speculative)
        // A fragments (8B aligned): taps {koff,koff+1}+{0,4,8}
        v2f a0 = *(const v2f*)(wc + 0);
        v2f a1 = *(const v2f*)(wc + 4);
        v2f a2 = *(const v2f*)(wc + 8);
        // B fragments (4B aligned pair loads)
        v2f b00 = *(const v2fu*)(i0 + koff + 0);
        v2f b01 = *(const v2fu*)(i0 + koff + 4);
        v2f b10 = *(const v2fu*)(i1 + koff + 0);
        v2f b11 = *(const v2fu*)(i1 + koff + 4);
        v2f t02 = *(const v2fu*)(i0 + 8);
        v2f t12 = *(const v2fu*)(i1 + 8);
        v2f b02 = half ? z2 : t02;   // taps {8,9,pad,pad}: padded lanes -> 0
        v2f b12 = half ? z2 : t12;
        acc0 = __builtin_amdgcn_wmma_f32_16x16x4_f32(false, a0, false, b00, (short)0, acc0, false, false);
        acc1 = __builtin_amdgcn_wmma_f32_16x16x4_f32(false, a0, false, b10, (short)0, acc1, false, false);
        acc0 = __builtin_amdgcn_wmma_f32_16x16x4_f32(false, a1, false, b01, (short)0, acc0, false, false);
        acc1 = __builtin_amdgcn_wmma_f32_16x16x4_f32(false, a1, false, b11, (short)0, acc1, false, false);
        acc0 = __builtin_amdgcn_wmma_f32_16x16x4_f32(false, a2, false, b02, (short)0, acc0, false, false);
        acc1 = __builtin_amdgcn_wmma_f32_16x16x4_f32(false, a2, false, b12, (short)0, acc1, false, false);
    }

    float* __restrict__ outb = out + (size_t)b * M * N;
    if (bcol0 < N) {
#pragma unroll
        for (int r = 0; r < 8; ++r) {
            const int m = m0 + half * 8 + r;
            if (m < M) outb[(size_t)m * N + bcol0] = acc0[r] + bias[m];
        }
    }
    if (bcol1 < N) {
#pragma unroll
        for (int r = 0; r < 8; ++r) {
            const int m = m0 + half * 8 + r;
            if (m < M) outb[(size_t)m * N + bcol1] = acc1[r] + bias[m];
        }
    }
}

// ---------------------------------------------------------------------------
// ELU monotone -> pool(elu(x)) == elu(max3(x)).  pre:[B,M,N] -> out:[B,M,Tout]
// ---------------------------------------------------------------------------
__global__ void elu_pool_kernel(const float* __restrict__ pre,
                                float* __restrict__ out,
                                int M, int N, int Tout, int total)
{
    int idx = blockIdx.x * blockDim.x + threadIdx.x;
    if (idx >= total) return;
    int t = idx % Tout;
    int m = (idx / Tout) % M;
    int b = idx / (Tout * M);
    const float* p = pre + ((size_t)b * M + m) * N + 3 * t;
    float v = fmaxf(p[0], fmaxf(p[1], p[2]));
    out[idx] = (v > 0.f) ? v : (__expf(v) - 1.f);
}

// ---------------------------------------------------------------------------
// Per-sample head: stage feat[b,:] into LDS with the CDNA5 async-to-LDS path
// (GLOBAL_LOAD_ASYNC_TO_LDS_B32, ASYNCcnt-tracked), then 4 waves compute the
// 4 outputs from LDS.  out[b,o] = dot(headsW[sid[b],o,:], feat[b,:]) + b
// ---------------------------------------------------------------------------
__global__ __launch_bounds__(128)
void head_kernel(const float* __restrict__ feat,   // [B,1400]
                 const float* __restrict__ hW,     // [K,4,1400]
                 const float* __restrict__ hB,     // [K,4]
                 const int* __restrict__ sid,      // [B]
                 float* __restrict__ out)          // [B,4]
{
    __shared__ float sfeat[1408];
    const int b = blockIdx.x;
    const int tid = threadIdx.x;
    const float* __restrict__ fv = feat + (size_t)b * 1400;

    // async copy global -> LDS (per-lane addresses; low 32b of flat shared ptr
    // is the wave-relative LDS byte offset per the LDS aperture rule)
    for (int i = tid; i < 1400; i += 128) {
        unsigned lds_off = (unsigned)(uintptr_t)(&sfeat[i]);
        const float* g = fv + i;
        asm volatile("global_load_async_to_lds_b32 %0, %1, off"
                     :: "v"(lds_off), "v"(g) : "memory");
    }
    asm volatile("s_wait_asynccnt 0" ::: "memory");
    __syncthreads();

    const int o = tid >> 5;
    const int lane = tid & 31;
    const int s = sid[b];
    const float* __restrict__ wv = hW + ((size_t)s * 4 + o) * 1400;
    float acc = 0.f;
    for (int i = lane; i < 1400; i += 32) acc += wv[i] * sfeat[i];
    for (int off = 16; off > 0; off >>= 1) acc += __shfl_down(acc, off, 32);
    if (lane == 0) out[b * 4 + o] = acc + hB[s * 4 + o];
}

// ---------------------------------------------------------------------------
extern "C" void kernel_launch(void* const* d_in, const int* in_sizes, int n_in,
                              void* d_out, int out_size, void* d_ws, size_t ws_size,
                              hipStream_t stream) {
    const float* x      = (const float*)d_in[0];   // [256,22,1000]
    const int*   sid    = (const int*)  d_in[1];   // [256]
    const float* w_time = (const float*)d_in[2];   // [25,1,10,1]
    const float* b_time = (const float*)d_in[3];
    const float* w_spat = (const float*)d_in[4];   // [25,25,1,22]
    const float* b_spat = (const float*)d_in[5];
    const float* w2     = (const float*)d_in[6];   // [50,25,10,1]
    const float* b2     = (const float*)d_in[7];
    const float* w3     = (const float*)d_in[8];   // [100,50,10,1]
    const float* b3     = (const float*)d_in[9];
    const float* w4     = (const float*)d_in[10];  // [200,100,10,1]
    const float* b4     = (const float*)d_in[11];
    const float* headsW = (const float*)d_in[12];  // [10,4,1400]
    const float* headsB = (const float*)d_in[13];  // [10,4]
    float* outp = (float*)d_out;                   // [256,4]

    const int B = 256;
    float* ws   = (float*)d_ws;
    float* wp1  = ws;                 // 32*264   =   8,448
    float* wp2  = ws + 8448;          // 64*300   =  19,200
    float* wp3  = ws + 27648;         // 112*600  =  67,200
    float* wp4  = ws + 94848;         // 208*1200 = 249,600
    float* beff = ws + 344448;        // 32
    float* pre  = ws + 345088;        // max 256*25*991 = 6,342,400
    float* actA = pre + 6400000;      // max(act1,act3) = 256*25*330 = 2,112,000
    float* actB = actA + 2112000;     // max(act2,feat) = 256*50*107 = 1,369,600

    // weight prep
    weff_kernel<<<(32 * 264 + 255) / 256, 256, 0, stream>>>(w_time, b_time, w_spat, b_spat, wp1, beff);
    repack_kernel<<<(64 * 300 + 255) / 256, 256, 0, stream>>>(w2, wp2, 50, 64, 25);
    repack_kernel<<<(112 * 600 + 255) / 256, 256, 0, stream>>>(w3, wp3, 100, 112, 50);
    repack_kernel<<<(208 * 1200 + 255) / 256, 256, 0, stream>>>(w4, wp4, 200, 208, 100);

    // layer 1 (fused time+spatial conv): M=25(pad32), K=220, N=991 -> pool 330
    gemm_conv_wmma<22, 1000, 25><<<dim3(31, 2, B), 32, 0, stream>>>(x, wp1, beff, pre);
    elu_pool_kernel<<<(B * 25 * 330 + 255) / 256, 256, 0, stream>>>(pre, actA, 25, 991, 330, B * 25 * 330);
    // layer 2: M=50(pad64), K=250, N=321 -> pool 107
    gemm_conv_wmma<25, 330, 50><<<dim3(11, 4, B), 32, 0, stream>>>(actA, wp2, b2, pre);
    elu_pool_kernel<<<(B * 50 * 107 + 255) / 256, 256, 0, stream>>>(pre, actB, 50, 321, 107, B * 50 * 107);
    // layer 3: M=100(pad112), K=500, N=98 -> pool 32
    gemm_conv_wmma<50, 107, 100><<<dim3(4, 7, B), 32, 0, stream>>>(actB, wp3, b3, pre);
    elu_pool_kernel<<<(B * 100 * 32 + 255) / 256, 256, 0, stream>>>(pre, actA, 100, 98, 32, B * 100 * 32);
    // layer 4: M=200(pad208), K=1000, N=23 -> pool 7
    gemm_conv_wmma<100, 32, 200><<<dim3(1, 13, B), 32, 0, stream>>>(actA, wp4, b4, pre);
    elu_pool_kernel<<<(B * 200 * 7 + 255) / 256, 256, 0, stream>>>(pre, actB, 200, 23, 7, B * 200 * 7);
    // heads; actB == feat [B,1400] (m*7+t flatten matches reference reshape)
    head_kernel<<<B, 128, 0, stream>>>(actB, headsW, headsB, sid, outp);
}